// FFT_Layer_60155311947872
// MI455X (gfx1250) — compile-verified
//
#include <hip/hip_runtime.h>
#include <math.h>

// ---------------------------------------------------------------------------
// FFT conv layer: out = IFFT2( (FFT2(imgs) contracted-over-Cin with W) )
// B=8, Cin=64, Cout=128, H=W=64.  Memory-bound (weight = 268MB read-once),
// so: fp32 everywhere, V_WMMA_F32_16X16X4_F32 for the per-frequency complex
// GEMM, LDS-tiled pair-interleaved weight transpose so every GEMM operand is
// a coalesced b64 load, naive DFT-64 (LDS twiddle table) for the FFT passes.
// ---------------------------------------------------------------------------

typedef __attribute__((ext_vector_type(2))) float v2f;
typedef __attribute__((ext_vector_type(8))) float v8f;

#define SDIM 64
#define NB   8
#define CIN  64
#define COUT 128
#define HW   (SDIM * SDIM)
#define PI_F 3.14159265358979323846f

// Build a 64-entry twiddle table in LDS: tw[t] = exp(sgn * i * 2*pi*t/64)
__device__ __forceinline__ void build_tw(float2* tw, float sgn) {
  int t = threadIdx.x;
  if (t < 64) {
    float ang = sgn * 2.0f * PI_F * (float)t * (1.0f / 64.0f);
    float s, c;
    __sincosf(ang, &s, &c);
    tw[t] = make_float2(c, s);
  }
}

__device__ __forceinline__ void dft64_accum(const float2* row, const float2* tw,
                                            int t, float& ar, float& ai) {
  ar = 0.0f; ai = 0.0f;
#pragma unroll 8
  for (int j = 0; j < 64; ++j) {
    float2 c = tw[(j * t) & 63];
    float2 x = row[j];
    ar = fmaf(x.x, c.x, fmaf(-x.y, c.y, ar));
    ai = fmaf(x.x, c.y, fmaf( x.y, c.x, ai));
  }
}

// K1: forward FFT along W.  One 64-thread block per (b,i,h) row. Coalesced.
__global__ void fft_w_kernel(const float* __restrict__ in, float* __restrict__ out) {
  __shared__ float2 tw[64];
  __shared__ float2 row[64];
  int t = threadIdx.x;
  build_tw(tw, -1.0f);
  size_t rid = blockIdx.x;                 // (b*CIN + i)*S + h
  row[t] = ((const float2*)in)[rid * SDIM + t];
  __syncthreads();
  float ar, ai;
  dft64_accum(row, tw, t, ar, ai);
  ((float2*)out)[rid * SDIM + t] = make_float2(ar, ai);
}

// K2: forward FFT along H + transpose to frequency-major A tiles.
// A layout: Aw[(f*16 + r)*64 + i], rows r=0..7 = Re(F[b]), r=8..15 = Im(F[b]).
// Applies the full ortho scale 1/64 for the forward FFT2.
__global__ void fft_h_kernel(const float* __restrict__ in, float* __restrict__ Aw) {
  __shared__ float2 tw[64];
  __shared__ float2 col[64];
  int t = threadIdx.x;
  build_tw(tw, -1.0f);
  int q = blockIdx.x;                      // (b, i, w)
  int b = q / (CIN * SDIM);
  int i = (q / SDIM) % CIN;
  int w = q % SDIM;
  size_t base = ((size_t)(b * CIN + i) * SDIM) * SDIM + w;  // complex index of (h=0,w)
  col[t] = ((const float2*)in)[base + (size_t)t * SDIM];
  __syncthreads();
  float ar, ai;
  dft64_accum(col, tw, t, ar, ai);
  const float sc = 1.0f / 64.0f;
  int f = t * SDIM + w;                    // frequency index h_out*64 + w
  Aw[((size_t)f * 16 + b)     * CIN + i] = ar * sc;
  Aw[((size_t)f * 16 + 8 + b) * CIN + i] = ai * sc;
}

// K3: weight transpose (i,o,f,part) -> Wt[((f*2+part)*32 + i/2)*256 + o*2 + (i&1)]
// i.e. frequency-major with (k,k+1) pairs interleaved per column, so each GEMM
// B-fragment is a single b64 load.  Block handles an i-pair so that both the
// read (along f) and the write (along o, float2) are fully coalesced via LDS.
__global__ void wtrans_kernel(const float* __restrict__ w, float* __restrict__ Wt) {
  __shared__ float2 tileA[16][17];
  __shared__ float2 tileB[16][17];
  int f0 = blockIdx.x * 16;
  int o0 = blockIdx.y * 16;
  int ip = blockIdx.z;                     // i-pair index, i = {2*ip, 2*ip+1}
  int tid = threadIdx.x;
  int tf = tid & 15, to = tid >> 4;
  tileA[to][tf] = ((const float2*)w)[(size_t)((2 * ip)     * COUT + (o0 + to)) * HW + (f0 + tf)];
  tileB[to][tf] = ((const float2*)w)[(size_t)((2 * ip + 1) * COUT + (o0 + to)) * HW + (f0 + tf)];
  __syncthreads();
  int to2 = tid & 15, tf2 = tid >> 4;
  float2 a = tileA[to2][tf2];              // i = 2*ip   : (Re, Im)
  float2 b = tileB[to2][tf2];              // i = 2*ip+1 : (Re, Im)
  size_t f = (size_t)(f0 + tf2);
  size_t o = (size_t)(o0 + to2);
  ((float2*)Wt)[((f * 2 + 0) * 32 + ip) * 128 + o] = make_float2(a.x, b.x);  // Re pair
  ((float2*)Wt)[((f * 2 + 1) * 32 + ip) * 128 + o] = make_float2(a.y, b.y);  // Im pair
}

// K4: per-frequency complex GEMM via fp32 WMMA.
// Block = 8 waves; wave handles N-tile o0 = wave*16.  A is 16x64 (Re rows 0-7,
// Im rows 8-15); two real accumulations A*Wr, A*Wi over 16 K-steps of 4.
// Combine: lane<16 holds row v (Fr*W), lane>=16 holds row v+8 (Fi*W); one
// xor-16 shuffle of ci gives every lane exactly the cross term it needs.
__global__ void __launch_bounds__(256) gemm_kernel(const float* __restrict__ Aw,
                                                   const float* __restrict__ Wt,
                                                   float* __restrict__ G) {
  int f    = blockIdx.x;
  int lane = threadIdx.x & 31;
  int wave = threadIdx.x >> 5;
  int o0   = wave * 16;

  const float*  Abase = Aw + (size_t)f * 16 * CIN;
  const float2* Wr2 = (const float2*)Wt + ((size_t)f * 2 + 0) * 32 * 128;
  const float2* Wi2 = (const float2*)Wt + ((size_t)f * 2 + 1) * 32 * 128;

  // Fragment lane mapping (ISA 7.12.2, 32-bit 16x4 A / 4x16 B):
  //   lanes 0-15: rows/cols 0-15, K = 4k+{0,1};  lanes 16-31: K = 4k+{2,3}
  int r    = lane & 15;
  int koff = (lane < 16) ? 0 : 2;
  int kph  = (lane < 16) ? 0 : 1;          // K-pair parity
  int nn   = o0 + (lane & 15);

  v8f cr = {};   // A x Wr  -> rows 0-7: Fr*Wr, rows 8-15: Fi*Wr
  v8f ci = {};   // A x Wi  -> rows 0-7: Fr*Wi, rows 8-15: Fi*Wi
  for (int kk = 0; kk < 16; ++kk) {
    int k  = kk * 4 + koff;
    int kp = kk * 2 + kph;
    float2 av = *(const float2*)(Abase + r * CIN + k);      // 8B aligned (k even)
    float2 bw = Wr2[(size_t)kp * 128 + nn];                 // (k, k+1) Re pair
    float2 bv = Wi2[(size_t)kp * 128 + nn];                 // (k, k+1) Im pair
    // Stream-prefetch the weight a few K-steps ahead (global_prefetch_b8).
    __builtin_prefetch((const void*)(Wr2 + (size_t)(kp + 8) * 128 + nn), 0, 1);
    __builtin_prefetch((const void*)(Wi2 + (size_t)(kp + 8) * 128 + nn), 0, 1);
    v2f a;  a.x  = av.x; a.y  = av.y;
    v2f br; br.x = bw.x; br.y = bw.y;
    v2f bi; bi.x = bv.x; bi.y = bv.y;
    cr = __builtin_amdgcn_wmma_f32_16x16x4_f32(false, a, false, br, (short)0, cr, false, false);
    ci = __builtin_amdgcn_wmma_f32_16x16x4_f32(false, a, false, bi, (short)0, ci, false, false);
  }

  // C/D layout: lane l, vgpr v -> (M = v + (l<16 ? 0 : 8), N = l&15).
  //   lane<16 : out_r(b=v)   = cr[v](Fr*Wr) - ci_sw(Fi*Wi)      row v
  //   lane>=16: out_i(b=v)   = cr[v](Fi*Wr) + ci_sw(Fr*Wi)      row v+8
  float  sgn     = (lane < 16) ? -1.0f : 1.0f;
  size_t rowbase = (size_t)f * 16 + ((lane < 16) ? 0 : 8);
#pragma unroll
  for (int v = 0; v < 8; ++v) {
    float ci_sw = __shfl_xor(ci[v], 16, 32);
    G[(rowbase + v) * COUT + nn] = fmaf(sgn, ci_sw, cr[v]);
  }
}

// K5: inverse FFT along W, reading frequency-major G, writing (b,o,h,w) complex.
__global__ void ifft_w_kernel(const float* __restrict__ G, float* __restrict__ Yb) {
  __shared__ float2 tw[64];
  __shared__ float2 row[64];
  int t = threadIdx.x;
  build_tw(tw, +1.0f);
  int q = blockIdx.x;                      // (b, o, h)
  int b = q / (COUT * SDIM);
  int o = (q / SDIM) % COUT;
  int h = q % SDIM;
  int fj = h * SDIM + t;
  float re = G[((size_t)fj * 16 + b)     * COUT + o];
  float im = G[((size_t)fj * 16 + 8 + b) * COUT + o];
  row[t] = make_float2(re, im);
  __syncthreads();
  float ar, ai;
  dft64_accum(row, tw, t, ar, ai);
  ((float2*)Yb)[((size_t)(b * COUT + o) * SDIM + h) * SDIM + t] = make_float2(ar, ai);
}

// K6: inverse FFT along H, ortho scale 1/64, final packed (…,2) output.
__global__ void ifft_h_kernel(const float* __restrict__ Yb, float* __restrict__ out) {
  __shared__ float2 tw[64];
  __shared__ float2 col[64];
  int t = threadIdx.x;
  build_tw(tw, +1.0f);
  int q = blockIdx.x;                      // (b, o, w)
  int b = q / (COUT * SDIM);
  int o = (q / SDIM) % COUT;
  int w = q % SDIM;
  size_t base = ((size_t)(b * COUT + o) * SDIM) * SDIM + w;  // complex idx (h=0,w)
  col[t] = ((const float2*)Yb)[base + (size_t)t * SDIM];
  __syncthreads();
  float ar, ai;
  dft64_accum(col, tw, t, ar, ai);
  const float sc = 1.0f / 64.0f;
  ((float2*)out)[base + (size_t)t * SDIM] = make_float2(ar * sc, ai * sc);
}

extern "C" void kernel_launch(void* const* d_in, const int* in_sizes, int n_in,
                              void* d_out, int out_size, void* d_ws, size_t ws_size,
                              hipStream_t stream) {
  const float* imgs   = (const float*)d_in[0];   // (8,64,64,64,2) f32
  const float* weight = (const float*)d_in[1];   // (1,64,128,64,64,2) f32
  float* out = (float*)d_out;                    // (8,128,64,64,2) f32

  // Workspace carve-up (floats). Yb aliases Wt (dead after the GEMM).
  float* Fa  = (float*)d_ws;                                   // 16 MB
  float* Awp = Fa  + (size_t)NB * CIN * HW * 2;                // 16 MB
  float* Wtp = Awp + (size_t)HW * 16 * CIN;                    // 268 MB
  float* Gp  = Wtp + (size_t)HW * 2 * CIN * COUT;              // 32 MB
  float* Ybp = Wtp;                                            // alias

  fft_w_kernel <<<NB * CIN * SDIM, 64, 0, stream>>>(imgs, Fa);
  fft_h_kernel <<<NB * CIN * SDIM, 64, 0, stream>>>(Fa, Awp);
  wtrans_kernel<<<dim3(HW / 16, COUT / 16, CIN / 2), 256, 0, stream>>>(weight, Wtp);
  gemm_kernel  <<<HW, 256, 0, stream>>>(Awp, Wtp, Gp);
  ifft_w_kernel<<<NB * COUT * SDIM, 64, 0, stream>>>(Gp, Ybp);
  ifft_h_kernel<<<NB * COUT * SDIM, 64, 0, stream>>>(Ybp, out);
}